// NeedlemanWunschAligner_57166014710128
// MI455X (gfx1250) — compile-verified
//
#include <hip/hip_runtime.h>
#include <hip/hip_bf16.h>

typedef __attribute__((ext_vector_type(2))) float v2f;
typedef __attribute__((ext_vector_type(8))) float v8f;

#define BATCH 8
#define NVAL  1024
#define DVAL  256
#define NEGV  (-1.0e10f)
#define VP    1026   // row pitch of the V scratch table (1025 cols + pad)

// ---------------------------------------------------------------------------
// C[b][i][j] = sum_d X[b][i][d] * Y[b][j][d]   (fp32, WMMA 16x16x4)
// One wave computes one 16x16 output tile; block = 8 waves = 128 rows x 16 cols.
// A-frag f32 16x4: lanes 0-15 hold (K=0,1) of row M=lane; lanes 16-31 hold (K=2,3).
// B-frag f32 4x16: symmetric, with N=lane&15 (Y is row-major [M][D], i.e. B^T).
// C layout: vgpr r -> row i0 + r + 8*(lane>=16), col j0 + (lane&15).
// ---------------------------------------------------------------------------
__global__ __launch_bounds__(256)
void gemm_nt_f32_wmma(const float* __restrict__ X, const float* __restrict__ Y,
                      float* __restrict__ Out) {
  const int lane = threadIdx.x & 31;
  const int wave = threadIdx.x >> 5;
  const int b    = blockIdx.z;
  const int j0   = blockIdx.x << 4;
  const int i0   = ((blockIdx.y << 3) + wave) << 4;
  const int half = lane >> 4;   // 0: lanes 0-15, 1: lanes 16-31
  const int l16  = lane & 15;

  const float* xrow = X + ((size_t)b * NVAL + (i0 + l16)) * DVAL + (half << 1);
  const float* yrow = Y + ((size_t)b * NVAL + (j0 + l16)) * DVAL + (half << 1);

  v8f c = {};
#pragma unroll 4
  for (int k = 0; k < DVAL; k += 4) {
    v2f a  = *(const v2f*)(xrow + k);
    v2f bb = *(const v2f*)(yrow + k);
    // D = A(16x4) * B(4x16) + C, all fp32
    c = __builtin_amdgcn_wmma_f32_16x16x4_f32(false, a, false, bb,
                                              (short)0, c, false, false);
  }

  float* o = Out + ((size_t)b * NVAL + (i0 + (half << 3))) * NVAL + j0 + l16;
#pragma unroll
  for (int r = 0; r < 8; ++r)
    o[(size_t)r * NVAL] = c[r];
}

// ---------------------------------------------------------------------------
// Forward smooth-NW DP. One workgroup per batch; anti-diagonal wavefront with
// the last three diagonals kept in LDS (critical chain never leaves the WGP).
// V is also streamed to a (N+1)x(M+1) global table (L2-resident) for backward.
//   V[i,j] = theta[i-1,j-1] + lse(A+V[i-1,j], V[i-1,j-1], A+V[i,j-1])
// ---------------------------------------------------------------------------
__global__ __launch_bounds__(1024)
void nw_forward(const float* __restrict__ theta, const float* __restrict__ A,
                float* __restrict__ Vg) {
  const int b = blockIdx.x;
  const float* th = theta + (size_t)b * NVAL * NVAL;
  const float* Ab = A     + (size_t)b * NVAL * NVAL;
  float* V = Vg + (size_t)b * (NVAL + 1) * VP;

  __shared__ float buf[3][1040];
  const int tid = threadIdx.x;

  for (int d = 0; d <= 2 * NVAL; ++d) {
    float*       cur = buf[d % 3];
    const float* dm1 = buf[(d + 2) % 3];   // diagonal d-1
    const float* dm2 = buf[(d + 1) % 3];   // diagonal d-2
    const int ilo = (d - NVAL) > 0 ? (d - NVAL) : 0;
    const int ihi = d < NVAL ? d : NVAL;
    for (int i = ilo + tid; i <= ihi; i += 1024) {
      const int j = d - i;
      float v;
      if (i == 0 && j == 0)      v = 0.0f;
      else if (i == 0 || j == 0) v = NEGV;
      else {
        const float a = Ab[(size_t)(i - 1) * NVAL + (j - 1)];
        const float t = th[(size_t)(i - 1) * NVAL + (j - 1)];
        const float x = a + dm1[i - 1];   // up:   A + V[i-1,j]
        const float y = dm2[i - 1];       // diag: V[i-1,j-1]
        const float z = a + dm1[i];       // left: A + V[i,j-1]
        const float m = fmaxf(fmaxf(x, y), z);
        v = t + m + __logf(__expf(x - m) + __expf(y - m) + __expf(z - m));
      }
      cur[i] = v;
      V[(size_t)i * VP + j] = v;
    }
    __syncthreads();
  }
}

// ---------------------------------------------------------------------------
// Backward (adjoint) DP: E[i,j] = dVt/dV[i,j]; aln[i-1,j-1] = E[i,j].
//   E[i,j] = E[i+1,j]*w_up(i+1,j) + E[i+1,j+1]*w_dg(i+1,j+1) + E[i,j+1]*w_lf(i,j+1)
// with softmax weights recomputed from V (exponents <= 0 by LSE property).
// ---------------------------------------------------------------------------
__global__ __launch_bounds__(1024)
void nw_backward(const float* __restrict__ theta, const float* __restrict__ A,
                 const float* __restrict__ Vg, float* __restrict__ aln) {
  const int b = blockIdx.x;
  const float* th = theta + (size_t)b * NVAL * NVAL;
  const float* Ab = A     + (size_t)b * NVAL * NVAL;
  const float* V  = Vg + (size_t)b * (NVAL + 1) * VP;
  float* out = aln + (size_t)b * NVAL * NVAL;

  __shared__ float buf[3][1040];
  const int tid = threadIdx.x;

  for (int d = 2 * NVAL; d >= 0; --d) {
    float*       cur = buf[d % 3];
    const float* dp1 = buf[(d + 1) % 3];   // diagonal d+1
    const float* dp2 = buf[(d + 2) % 3];   // diagonal d+2
    const int ilo = (d - NVAL) > 0 ? (d - NVAL) : 0;
    const int ihi = d < NVAL ? d : NVAL;
    for (int i = ilo + tid; i <= ihi; i += 1024) {
      const int j = d - i;
      float e;
      if (i == NVAL && j == NVAL) {
        e = 1.0f;
      } else {
        const float vij = V[(size_t)i * VP + j];
        e = 0.0f;
        if (i + 1 <= NVAL && j >= 1) {        // successor (i+1,j), "up" branch
          const float w = __expf(Ab[(size_t)i * NVAL + (j - 1)] + vij +
                                 th[(size_t)i * NVAL + (j - 1)] -
                                 V[(size_t)(i + 1) * VP + j]);
          e += dp1[i + 1] * w;
        }
        if (i + 1 <= NVAL && j + 1 <= NVAL) { // successor (i+1,j+1), "diag"
          const float w = __expf(vij + th[(size_t)i * NVAL + j] -
                                 V[(size_t)(i + 1) * VP + (j + 1)]);
          e += dp2[i + 1] * w;
        }
        if (j + 1 <= NVAL && i >= 1) {        // successor (i,j+1), "left"
          const float w = __expf(Ab[(size_t)(i - 1) * NVAL + j] + vij +
                                 th[(size_t)(i - 1) * NVAL + j] -
                                 V[(size_t)i * VP + (j + 1)]);
          e += dp1[i] * w;
        }
      }
      cur[i] = e;
      if (i >= 1 && j >= 1)
        out[(size_t)(i - 1) * NVAL + (j - 1)] = e;
    }
    __syncthreads();
  }
}

// ---------------------------------------------------------------------------
extern "C" void kernel_launch(void* const* d_in, const int* in_sizes, int n_in,
                              void* d_out, int out_size, void* d_ws, size_t ws_size,
                              hipStream_t stream) {
  const float* zx = (const float*)d_in[0];
  const float* zy = (const float*)d_in[1];
  const float* gx = (const float*)d_in[2];
  const float* gy = (const float*)d_in[3];

  float* aln   = (float*)d_out;                         // [8,1024,1024]
  float* theta = aln   + (size_t)BATCH * NVAL * NVAL;   // [8,1024,1024]
  float* Amat  = theta + (size_t)BATCH * NVAL * NVAL;   // [8,1024,1024]
  float* Vg    = (float*)d_ws;                          // [8,1025,VP] ~33.7 MB

  dim3 gg(NVAL / 16, NVAL / 16 / 8, BATCH);             // (64, 8, 8)
  gemm_nt_f32_wmma<<<gg, 256, 0, stream>>>(zx, zy, theta);
  gemm_nt_f32_wmma<<<gg, 256, 0, stream>>>(gx, gy, Amat);

  nw_forward <<<BATCH, 1024, 0, stream>>>(theta, Amat, Vg);
  nw_backward<<<BATCH, 1024, 0, stream>>>(theta, Amat, Vg, aln);
}